// LSTMModel_19662360281090
// MI455X (gfx1250) — compile-verified
//
#include <hip/hip_runtime.h>
#include <hip/hip_bf16.h>

typedef float v2f __attribute__((ext_vector_type(2)));
typedef float v4f __attribute__((ext_vector_type(4)));
typedef float v8f __attribute__((ext_vector_type(8)));

#define T_TOTAL 512
#define BATCH   4096
#define DIN     32

__device__ __forceinline__ float sigf(float x) {
    return 1.0f / (1.0f + __expf(-x));
}
// tanh via exp of non-positive argument only (no overflow -> no NaN)
__device__ __forceinline__ float tanhx(float x) {
    float t = __expf(-2.0f * fabsf(x));
    float r = (1.0f - t) / (1.0f + t);
    return copysignf(r, x);
}

// ---------------------------------------------------------------------------
// Phase 1: G[t,b,g] = sum_k x[t,b,k] * Wih1[g,k] + (bih1[g]+bhh1[g])
// One wave per (t, 16-batch tile). WMMA f32 16x16x4:
//   A (16x4, M=gate padded 8->16, K)   : from Wih1, loaded once per wave
//   B (4x16,  K, N=batch)              : from x, b64 per lane per K-chunk
//   D (16x16, M=gate, N=batch)         : lane n (n<16) holds all 8 gates of
//                                        batch b0+n in VGPRs 0..7 -> 2x b128 store
// ---------------------------------------------------------------------------
__global__ __launch_bounds__(256) void lstm1_gates_wmma(
    const float* __restrict__ x,      // (T, B, D)
    const float* __restrict__ Wih1,   // (8, 32)
    const float* __restrict__ bih1,   // (8)
    const float* __restrict__ bhh1,   // (8)
    float* __restrict__ G,            // (tc, B, 8)
    int t0, int tc)
{
    const int lane  = threadIdx.x & 31;
    const int wave  = threadIdx.x >> 5;
    const int tile  = blockIdx.x * 8 + wave;   // 0 .. tc*(B/16)-1
    const int trel  = tile >> 8;               // tile / (BATCH/16)
    const int b0    = (tile & 255) << 4;
    if (trel >= tc) return;                    // wave-uniform guard
    const int m     = lane & 15;               // M row (A) / N col (B,D)
    const int khalf = lane >> 4;               // K half selector

    // A fragments: Wih1 as 16x4 tiles (rows 8..15 are zero padding)
    v2f a[8];
#pragma unroll
    for (int kc = 0; kc < 8; ++kc) {
        const int k = kc * 4 + khalf * 2;
        float a0 = 0.0f, a1 = 0.0f;
        if (m < 8) { a0 = Wih1[m * DIN + k]; a1 = Wih1[m * DIN + k + 1]; }
        v2f av; av.x = a0; av.y = a1;
        a[kc] = av;
    }

    // C init with bias on the valid gate rows (M = 0..7 live in lane group 0)
    v8f c;
#pragma unroll
    for (int g = 0; g < 8; ++g)
        c[g] = (khalf == 0) ? (bih1[g] + bhh1[g]) : 0.0f;

    // B fragments from x: B[k][n] = x[t, b0+n, k]
    const float* xb = x + ((size_t)(t0 + trel) * BATCH + b0 + m) * DIN + khalf * 2;
#pragma unroll
    for (int kc = 0; kc < 8; ++kc) {
        v2f bf = *(const v2f*)(xb + kc * 4);   // b64, zero overfetch per wave
        c = __builtin_amdgcn_wmma_f32_16x16x4_f32(
                /*neg_a=*/false, a[kc], /*neg_b=*/false, bf,
                /*c_mod=*/(short)0, c, /*reuse_a=*/false, /*reuse_b=*/false);
    }

    // Lane n (group 0) holds gates 0..7 of batch b0+n -> contiguous 32B store
    if (khalf == 0) {
        float* gp = G + ((size_t)trel * BATCH + b0 + m) * 8;
        v4f lo; lo[0] = c[0]; lo[1] = c[1]; lo[2] = c[2]; lo[3] = c[3];
        v4f hi; hi[0] = c[4]; hi[1] = c[5]; hi[2] = c[6]; hi[3] = c[7];
        *(v4f*)(gp)     = lo;
        *(v4f*)(gp + 4) = hi;
    }
}

// ---------------------------------------------------------------------------
// Phase 2: sequential recurrence, one lane per batch element.
// Reads precomputed gates (32B/lane/step, L2-resident by chunking),
// carries h/c state across chunk launches via `state` in workspace.
// Gate order (PyTorch): [i0,i1, f0,f1, g0,g1, o0,o1].
// ---------------------------------------------------------------------------
__global__ __launch_bounds__(256) void lstm_recurrence(
    const float* __restrict__ G,      // (tc, B, 8)
    const float* __restrict__ Whh1,   // (8, 2)
    const float* __restrict__ Wih2,   // (4, 2)
    const float* __restrict__ Whh2,   // (4, 1)
    const float* __restrict__ bih2,   // (4)
    const float* __restrict__ bhh2,   // (4)
    float* __restrict__ state,        // 6 * B floats
    float* __restrict__ out,          // (T, B)
    int t0, int tc)
{
    const int b = blockIdx.x * blockDim.x + threadIdx.x;   // 0..BATCH-1

    float whh1[16];
#pragma unroll
    for (int j = 0; j < 16; ++j) whh1[j] = Whh1[j];
    float wih2[8];
#pragma unroll
    for (int j = 0; j < 8; ++j) wih2[j] = Wih2[j];
    float whh2v[4], b2[4];
#pragma unroll
    for (int j = 0; j < 4; ++j) { whh2v[j] = Whh2[j]; b2[j] = bih2[j] + bhh2[j]; }

    float h10, h11, c10, c11, h2, c2;
    if (t0 == 0) {
        h10 = h11 = c10 = c11 = h2 = c2 = 0.0f;
    } else {
        h10 = state[b];
        h11 = state[BATCH + b];
        c10 = state[2 * BATCH + b];
        c11 = state[3 * BATCH + b];
        h2  = state[4 * BATCH + b];
        c2  = state[5 * BATCH + b];
    }

    for (int t = 0; t < tc; ++t) {
        const float* gp = G + ((size_t)t * BATCH + b) * 8;
        v4f ga = *(const v4f*)(gp);
        v4f gb = *(const v4f*)(gp + 4);

        float g0 = ga[0] + h10 * whh1[0]  + h11 * whh1[1];
        float g1 = ga[1] + h10 * whh1[2]  + h11 * whh1[3];
        float g2 = ga[2] + h10 * whh1[4]  + h11 * whh1[5];
        float g3 = ga[3] + h10 * whh1[6]  + h11 * whh1[7];
        float g4 = gb[0] + h10 * whh1[8]  + h11 * whh1[9];
        float g5 = gb[1] + h10 * whh1[10] + h11 * whh1[11];
        float g6 = gb[2] + h10 * whh1[12] + h11 * whh1[13];
        float g7 = gb[3] + h10 * whh1[14] + h11 * whh1[15];

        float i0 = sigf(g0), i1 = sigf(g1);
        float f0 = sigf(g2), f1 = sigf(g3);
        float cc0 = tanhx(g4), cc1 = tanhx(g5);
        float o0 = sigf(g6), o1 = sigf(g7);
        c10 = f0 * c10 + i0 * cc0;
        c11 = f1 * c11 + i1 * cc1;
        h10 = o0 * tanhx(c10);
        h11 = o1 * tanhx(c11);

        float q0 = wih2[0] * h10 + wih2[1] * h11 + whh2v[0] * h2 + b2[0];
        float q1 = wih2[2] * h10 + wih2[3] * h11 + whh2v[1] * h2 + b2[1];
        float q2 = wih2[4] * h10 + wih2[5] * h11 + whh2v[2] * h2 + b2[2];
        float q3 = wih2[6] * h10 + wih2[7] * h11 + whh2v[3] * h2 + b2[3];
        float i2 = sigf(q0), f2 = sigf(q1), gq = tanhx(q2), o2 = sigf(q3);
        c2 = f2 * c2 + i2 * gq;
        h2 = o2 * tanhx(c2);

        out[(size_t)(t0 + t) * BATCH + b] = tanhx(h2);
    }

    state[b]             = h10;
    state[BATCH + b]     = h11;
    state[2 * BATCH + b] = c10;
    state[3 * BATCH + b] = c11;
    state[4 * BATCH + b] = h2;
    state[5 * BATCH + b] = c2;
}

extern "C" void kernel_launch(void* const* d_in, const int* in_sizes, int n_in,
                              void* d_out, int out_size, void* d_ws, size_t ws_size,
                              hipStream_t stream) {
    const float* x    = (const float*)d_in[0];
    const float* Wih1 = (const float*)d_in[1];
    const float* Whh1 = (const float*)d_in[2];
    const float* bih1 = (const float*)d_in[3];
    const float* bhh1 = (const float*)d_in[4];
    const float* Wih2 = (const float*)d_in[5];
    const float* Whh2 = (const float*)d_in[6];
    const float* bih2 = (const float*)d_in[7];
    const float* bhh2 = (const float*)d_in[8];
    float* out = (float*)d_out;

    // Workspace layout: [state: 6*B floats][G: tcmax*B*8 floats]
    float* state = (float*)d_ws;
    float* G     = state + 6 * BATCH;
    const size_t state_bytes = (size_t)6 * BATCH * sizeof(float);
    size_t avail = ws_size > state_bytes ? ws_size - state_bytes : 0;
    int tcmax = (int)(avail / ((size_t)BATCH * 8 * sizeof(float)));
    if (tcmax < 1)       tcmax = 1;        // assume ws is at least ~224 KB
    if (tcmax > T_TOTAL) tcmax = T_TOTAL;

    for (int t0 = 0; t0 < T_TOTAL; t0 += tcmax) {
        const int tc = (T_TOTAL - t0 < tcmax) ? (T_TOTAL - t0) : tcmax;
        // tc * (B/16) waves, 8 waves (256 threads) per block -> tc*32 blocks
        lstm1_gates_wmma<<<tc * 32, 256, 0, stream>>>(x, Wih1, bih1, bhh1, G, t0, tc);
        lstm_recurrence<<<BATCH / 256, 256, 0, stream>>>(G, Whh1, Wih2, Whh2,
                                                         bih2, bhh2, state, out, t0, tc);
    }
    (void)in_sizes; (void)n_in; (void)out_size;
}